// DRR_71279277245091
// MI455X (gfx1250) — compile-verified
//
#include <hip/hip_runtime.h>

// DRR forward projector for MI455X (gfx1250).
//
// Roofline: 131072 rays x <=128 samples; 64MB f32 volume fully L2-resident
// (192MB L2) -> bound by L2 gather issue/latency and inner-loop VALU issue.
//  * per-ray slab clipping: only ~33/128 samples intersect the volume
//    (ray spans 1020mm, volume 256mm deep) -> ~4x less work. In-loop mask
//    retained so boundary semantics match the reference exactly.
//  * 16x16 pixel tile blocks -> compact, reused gather footprint in L0/L2.
//  * z-pair b64 gathers (4 loads/sample sharing one base addr + imm offsets).
//  * packed-f32 lerp tree (v_pk_fma_f32) + v_med3_f32 clamps to cut VALU.
//  * global_prefetch_b8 look-ahead along z (+~8 voxels = 32B per sample).
//  * clipped range split across SPLIT blocks (balanced), deterministic reduce.
// WMMA intentionally unused: no dense matrix structure, ~0.5 GFLOP total.

#define DVV   256   // volume side
#define HH    256   // detector height
#define WW    256   // detector width
#define BB    2     // batch
#define TILE  16    // pixel tile side (16x16 = 256 threads)
#define SPLIT 4     // live-range split for occupancy / balance

typedef float v2f __attribute__((ext_vector_type(2)));

// 8-byte z-pair with only 4-byte alignment guarantee.
struct __attribute__((packed, aligned(4))) ZPair { float lo, hi; };

__device__ __forceinline__ void slab_axis(float p, float s, float hif,
                                          float& lo, float& hi)
{
    if (fabsf(s) > 1e-12f) {
        const float a = (0.0f - p) / s;
        const float b = (hif - p) / s;
        lo = fminf(a, b);
        hi = fmaxf(a, b);
    } else {
        const bool in = (p >= 0.0f) && (p <= hif);
        lo = in ? -3.0e38f :  3.0e38f;
        hi = in ?  3.0e38f : -3.0e38f;
    }
}

__global__ __launch_bounds__(TILE * TILE)
void drr_pass1(const float* __restrict__ src,      // (B, H*W, 3)
               const float* __restrict__ tgt,      // (B, H*W, 3)
               const float* __restrict__ vol,      // (DV, DV, DV)
               const float* __restrict__ spacing,  // (3,)
               const float* __restrict__ origin,   // (3,)
               const int*   __restrict__ nptr,     // n_points
               float*       __restrict__ dst,      // (nsplit*B, H*W) partials
               int nsplit)
{
    const int tx  = threadIdx.x & (TILE - 1);
    const int ty  = threadIdx.x / TILE;
    const int px  = blockIdx.x * TILE + tx;
    const int py  = blockIdx.y * TILE + ty;
    const int b   = blockIdx.z % BB;
    const int sp  = blockIdx.z / BB;
    const int pix = py * WW + px;
    const int n   = *nptr;                         // uniform scalar load

    const int gbase = (b * (HH * WW) + pix) * 3;
    const float sx = src[gbase + 0], sy = src[gbase + 1], sz = src[gbase + 2];
    const float rx = tgt[gbase + 0] - sx;
    const float ry = tgt[gbase + 1] - sy;
    const float rz = tgt[gbase + 2] - sz;

    const float isx = 1.0f / spacing[0];
    const float isy = 1.0f / spacing[1];
    const float isz = 1.0f / spacing[2];
    const float p0x = (sx - origin[0]) * isx;      // voxel-space ray origin
    const float p0y = (sy - origin[1]) * isy;
    const float p0z = (sz - origin[2]) * isz;
    const float dt  = 1.0f / (float)(n - 1);       // linspace(0,1,n) step
    const float stx = rx * isx * dt;               // voxel-space step / sample
    const float sty = ry * isy * dt;
    const float stz = rz * isz * dt;

    const float hif = (float)(DVV - 1);

    // ---- per-ray slab clip: fx(i) = p0 + i*st is affine in sample index i.
    // Widen by 1-2 samples and keep the in-loop mask, so fp slop here can
    // never change the result vs. the reference's per-sample inside test.
    float lx, hx, ly, hy, lz, hz;
    slab_axis(p0x, stx, hif, lx, hx);
    slab_axis(p0y, sty, hif, ly, hy);
    slab_axis(p0z, stz, hif, lz, hz);
    float ilo = fmaxf(lx, fmaxf(ly, lz));
    float ihi = fminf(hx, fminf(hy, hz));
    ilo = fminf(fmaxf(ilo, 0.0f),  (float)n);      // clamp before int convert
    ihi = fminf(fmaxf(ihi, -1.0f), (float)n);
    const int iA0 = max(0, (int)floorf(ilo) - 1);
    const int iB0 = min(n, (int)ceilf(ihi) + 2);
    const int len = max(0, iB0 - iA0);

    // Partition the LIVE range across the split blocks (balanced; all split
    // blocks compute identical iA0/iB0 -> deterministic, gap/overlap free).
    const int chunk = (len + nsplit - 1) / nsplit;
    const int s0 = iA0 + sp * chunk;
    const int s1 = min(iB0, s0 + chunk);

    // z advances ~8 voxels (32B) per sample; prefetch ~2 samples ahead.
    const int pfdz = (int)(2.0f * stz);
    const float clampHi = (float)(DVV - 2);

    float acc = 0.0f;
    #pragma unroll 4
    for (int i = s0; i < s1; ++i) {
        const float fi = (float)i;
        const float fx = fmaf(fi, stx, p0x);
        const float fy = fmaf(fi, sty, p0y);
        const float fz = fmaf(fi, stz, p0z);

        // reference: inside = all(idx >= 0 & idx <= D-1)
        const float m = (fx >= 0.0f && fx <= hif &&
                         fy >= 0.0f && fy <= hif &&
                         fz >= 0.0f && fz <= hif) ? 1.0f : 0.0f;

        // Float-domain clamp of the base corner to [0, D-2] via v_med3_f32.
        // Fractional weight reaches 1.0 at the upper boundary, matching the
        // reference's clip(fi+di)+weight math for every inside point.
        // Clamping also keeps masked (outside) loads in-bounds.
        const float fcx = __builtin_amdgcn_fmed3f(floorf(fx), 0.0f, clampHi);
        const float fcy = __builtin_amdgcn_fmed3f(floorf(fy), 0.0f, clampHi);
        const float fcz = __builtin_amdgcn_fmed3f(floorf(fz), 0.0f, clampHi);
        const int cx = (int)fcx, cy = (int)fcy, cz = (int)fcz;
        const float wx = fx - fcx;
        const float wy = fy - fcy;
        const float wz = fz - fcz;

        // flat = (x*DV + y)*DV + z ; z contiguous -> 4x 8B loads sharing one
        // base address with immediate offsets (+1KB, +256KB, +257KB).
        const int r00 = (cx * DVV + cy) * DVV + cz;
        const ZPair z00 = *reinterpret_cast<const ZPair*>(vol + r00);
        const ZPair z01 = *reinterpret_cast<const ZPair*>(vol + r00 + DVV);
        const ZPair z10 = *reinterpret_cast<const ZPair*>(vol + r00 + DVV * DVV);
        const ZPair z11 = *reinterpret_cast<const ZPair*>(vol + r00 + DVV * DVV + DVV);

        // gfx1250 prefetch: same (x,y) row, z two samples ahead
        {
            const int zp = min(cz + pfdz, DVV - 2);
            __builtin_prefetch((const void*)(vol + (r00 - cz + zp)), 0, 3);
        }

        // Packed-f32 lerp tree: lerp y, then x, on the {z, z+1} lanes
        // (v_pk_add/v_pk_fma, each doing 2 lanes), final z-lerp scalar.
        // Same product-of-weights sum as the reference, reordered.
        const v2f a00 = { z00.lo, z00.hi };
        const v2f a01 = { z01.lo, z01.hi };
        const v2f a10 = { z10.lo, z10.hi };
        const v2f a11 = { z11.lo, z11.hi };
        const v2f wy2 = { wy, wy };
        const v2f wx2 = { wx, wx };
        const v2f py0 = __builtin_elementwise_fma(wy2, a01 - a00, a00);
        const v2f py1 = __builtin_elementwise_fma(wy2, a11 - a10, a10);
        const v2f qx  = __builtin_elementwise_fma(wx2, py1 - py0, py0);
        const float v = fmaf(wz, qx.y - qx.x, qx.x);

        acc = fmaf(v, m, acc);
    }

    const float raylen = sqrtf(fmaf(rx, rx, fmaf(ry, ry, rz * rz)));
    // partial already carries the raylen/n scaling -> reduce is a plain sum
    dst[(sp * BB + b) * (HH * WW) + pix] = acc * raylen / (float)n;
}

__global__ __launch_bounds__(256)
void drr_reduce(const float* __restrict__ ws, float* __restrict__ out)
{
    const int Npx = BB * HH * WW;
    const int i = blockIdx.x * blockDim.x + threadIdx.x;
    if (i < Npx) {
        float s = 0.0f;
        #pragma unroll
        for (int k = 0; k < SPLIT; ++k)
            s += ws[k * Npx + i];   // fixed order -> bit-deterministic
        out[i] = s;
    }
}

extern "C" void kernel_launch(void* const* d_in, const int* in_sizes, int n_in,
                              void* d_out, int out_size, void* d_ws, size_t ws_size,
                              hipStream_t stream)
{
    (void)in_sizes; (void)n_in; (void)out_size;
    const float* src     = (const float*)d_in[0];  // source  (B,H*W,3) f32
    const float* tgt     = (const float*)d_in[1];  // target  (B,H*W,3) f32
    const float* vol     = (const float*)d_in[2];  // density (DV^3)    f32
    const float* spacing = (const float*)d_in[3];  // (3,) f32
    const float* origin  = (const float*)d_in[4];  // (3,) f32
    const int*   nptr    = (const int*)d_in[5];    // n_points scalar

    const dim3 blk(TILE * TILE, 1, 1);
    const size_t needed = (size_t)SPLIT * BB * HH * WW * sizeof(float);

    if (ws_size >= needed) {
        // two-pass: live-range split for occupancy, deterministic reduce
        const dim3 grd(WW / TILE, HH / TILE, BB * SPLIT);
        drr_pass1<<<grd, blk, 0, stream>>>(src, tgt, vol, spacing, origin,
                                           nptr, (float*)d_ws, SPLIT);
        const int Npx = BB * HH * WW;
        drr_reduce<<<dim3((Npx + 255) / 256, 1, 1), dim3(256, 1, 1), 0, stream>>>(
            (const float*)d_ws, (float*)d_out);
    } else {
        // fallback: single pass straight to output
        const dim3 grd(WW / TILE, HH / TILE, BB);
        drr_pass1<<<grd, blk, 0, stream>>>(src, tgt, vol, spacing, origin,
                                           nptr, (float*)d_out, 1);
    }
}